// AnovaKernel_12421045420596
// MI455X (gfx1250) — compile-verified
//
#include <hip/hip_runtime.h>

typedef __attribute__((ext_vector_type(2))) float v2f;
typedef __attribute__((ext_vector_type(4))) float v4f;
typedef __attribute__((ext_vector_type(8))) float v8f;

namespace {
constexpr int kF = 64;          // feature (scan) axis
constexpr int kD = 64;          // lane axis (reduced at the end)
constexpr int kBPerBlock = 16;  // batch elements per block
constexpr int kThreads = 256;   // 8 wave32
}

// Per thread: 4 consecutive d's of one batch element b.
// 16 threads / b, 2 b's / wave, 16 b's / block.
// Streaming non-temporal b128 loads + register-resident order-3 scan,
// then a V_WMMA_F32_16X16X4_F32 reduction of the 16x16 partial tile.
__global__ __launch_bounds__(kThreads) void anova3_scan_wmma(
    const float* __restrict__ x, float* __restrict__ out, int B) {
  __shared__ float T[kBPerBlock][17];  // padded: 16 partials per b

  const int tid  = threadIdx.x;
  const int wave = tid >> 5;        // 0..7
  const int lane = tid & 31;
  const int half = lane >> 4;       // 0 or 1 (which b within the wave)
  const int l16  = lane & 15;       // d-chunk index / row index
  const int bLoc = (wave << 1) | half;               // 0..15
  const int b    = blockIdx.x * kBPerBlock + bLoc;

  v4f a1 = {0.f, 0.f, 0.f, 0.f};
  v4f a2 = {0.f, 0.f, 0.f, 0.f};
  v4f a3 = {0.f, 0.f, 0.f, 0.f};

  if (b < B) {
    const v4f* xp =
        reinterpret_cast<const v4f*>(x + (size_t)b * kF * kD) + l16;
#pragma unroll 8
    for (int f = 0; f < kF; ++f) {
      // read-once stream: non-temporal hint (TH=NT), 256B stride -> immediate offsets
      v4f v = __builtin_nontemporal_load(xp + f * (kD / 4));
      a3.x = fmaf(v.x, a2.x, a3.x); a2.x = fmaf(v.x, a1.x, a2.x); a1.x += v.x;
      a3.y = fmaf(v.y, a2.y, a3.y); a2.y = fmaf(v.y, a1.y, a2.y); a1.y += v.y;
      a3.z = fmaf(v.z, a2.z, a3.z); a2.z = fmaf(v.z, a1.z, a2.z); a1.z += v.z;
      a3.w = fmaf(v.w, a2.w, a3.w); a2.w = fmaf(v.w, a1.w, a2.w); a1.w += v.w;
    }
  }

  // per-thread partial over its 4 d's
  T[bLoc][l16] = (a3.x + a3.y) + (a3.z + a3.w);
  __syncthreads();

  if (wave == 0) {
    // Reduce each row of the 16x16 tile with 4 accumulating f32 WMMAs:
    // D = sum_j A_j(16x4 slice of T) x ones(4x16), so D[M][*] = sum_col T[M][col].
    v8f c = {0.f, 0.f, 0.f, 0.f, 0.f, 0.f, 0.f, 0.f};
    const v2f ones = {1.f, 1.f};
#pragma unroll
    for (int j = 0; j < 4; ++j) {
      // A 16x4 f32 layout: lane m -> A[m][0],A[m][1]; lane 16+m -> A[m][2],A[m][3]
      v2f a;
      a.x = T[l16][(j << 2) + (half << 1) + 0];
      a.y = T[l16][(j << 2) + (half << 1) + 1];
      c = __builtin_amdgcn_wmma_f32_16x16x4_f32(
          /*neg_a=*/false, a, /*neg_b=*/false, ones,
          /*c_mod=*/(short)0, c, /*reuse_a=*/false, /*reuse_b=*/false);
    }
    // C/D layout: VGPR r, lane 0 -> D[r][0] (b rows 0..7), lane 16 -> D[8+r][0]
    if (l16 == 0) {
      const int base = blockIdx.x * kBPerBlock + (half << 3);
      if (base + 8 <= B) {
        // fast path: two b128 stores of 8 contiguous outputs
        v4f lo = {c[0], c[1], c[2], c[3]};
        v4f hi = {c[4], c[5], c[6], c[7]};
        *reinterpret_cast<v4f*>(out + base)     = lo;
        *reinterpret_cast<v4f*>(out + base + 4) = hi;
      } else {
#pragma unroll
        for (int r = 0; r < 8; ++r) {
          if (base + r < B) out[base + r] = c[r];
        }
      }
    }
  }
}

extern "C" void kernel_launch(void* const* d_in, const int* in_sizes, int n_in,
                              void* d_out, int out_size, void* d_ws, size_t ws_size,
                              hipStream_t stream) {
  const float* x = (const float*)d_in[0];
  float* out = (float*)d_out;
  const int B = in_sizes[0] / (kF * kD);               // 8192 for the reference shape
  const int grid = (B + kBPerBlock - 1) / kBPerBlock;  // 512 blocks, 4096 waves
  anova3_scan_wmma<<<grid, kThreads, 0, stream>>>(x, out, B);
}